// RelationWiseWordSelectionHead_69836168233629
// MI455X (gfx1250) — compile-verified
//
#include <hip/hip_runtime.h>
#include <math.h>

typedef __attribute__((ext_vector_type(2))) float v2f;
typedef __attribute__((ext_vector_type(8))) float v8f;

#define BB 2
#define SS 192
#define HH 768
#define LL 4
#define NN (HH * LL)   // 3072
#define MM (BB * SS)   // 384

// ---------------------------------------------------------------------------
// Kernel 1: Y[m, n] = sum_k X[m,k] * W[n,k] + bias[n]
//   X: (384, 768)  W: (3072, 768)  Y: (384, 3072)
// One wave (32 lanes) computes a 16x16 tile of Y with V_WMMA_F32_16X16X4_F32.
// f32 A 16x4 layout: lane -> M = lane%16 ; VGPR v holds K = v + 2*(lane/16).
// f32 B 4x16 layout: lane -> N = lane%16 ; VGPR v holds K = v + 2*(lane/16).
// f32 C/D 16x16:     VGPR r, lane -> M = r + 8*(lane/16), N = lane%16.
// ---------------------------------------------------------------------------
__global__ __launch_bounds__(32) void proj_wmma_f32(
    const float* __restrict__ X, const float* __restrict__ W,
    const float* __restrict__ bias, float* __restrict__ Y) {
  const int n0 = blockIdx.x * 16;
  const int m0 = blockIdx.y * 16;
  const int lane = threadIdx.x;
  const int r16  = lane & 15;
  const int kh   = (lane >> 4) * 2;  // 0 or 2: K offset for this lane-half

  const float* __restrict__ arow = X + (size_t)(m0 + r16) * HH;
  const float* __restrict__ brow = W + (size_t)(n0 + r16) * HH;

  v8f acc = {};
  for (int k = 0; k < HH; k += 4) {
    v2f a = *(const v2f*)(arow + k + kh);  // A[m, k+kh], A[m, k+kh+1]
    v2f b = *(const v2f*)(brow + k + kh);  // B[k+kh, n] = W[n, k+kh], ...
    acc = __builtin_amdgcn_wmma_f32_16x16x4_f32(
        /*neg_a=*/false, a, /*neg_b=*/false, b,
        /*c_mod=*/(short)0, acc, /*reuse_a=*/false, /*reuse_b=*/false);
  }

  const int n     = n0 + r16;
  const int mbase = m0 + ((lane >> 4) << 3);
  const float bn  = bias[n];
#pragma unroll
  for (int r = 0; r < 8; ++r) {
    Y[(size_t)(mbase + r) * NN + n] = acc[r] + bn;
  }
}

// ---------------------------------------------------------------------------
// Kernel 2: out[b,s,t,l] = sum_h gelu(hsrc[b,s,l,h] + htgt[b,t,l,h]) * w[h,l]
// Block = (b,s,l): src row + classifier column staged in LDS (reused by all
// 192 t). 8 waves stride over t; each lane strides h by 32; exact erf GELU;
// wave32 shfl_xor reduction; lane 0 stores.
// ---------------------------------------------------------------------------
__global__ __launch_bounds__(256) void pair_gelu_reduce(
    const float* __restrict__ hsrc, const float* __restrict__ htgt,
    const float* __restrict__ wcls, float* __restrict__ out) {
  __shared__ float s_src[HH];
  __shared__ float s_w[HH];

  const int s = blockIdx.x;
  const int l = blockIdx.y;
  const int b = blockIdx.z;
  const int tid = threadIdx.x;

  const float* __restrict__ srcrow =
      hsrc + ((size_t)((b * SS + s) * LL + l)) * HH;
  for (int i = tid; i < HH; i += 256) {
    s_src[i] = srcrow[i];
    s_w[i]   = wcls[i * LL + l];
  }
  __syncthreads();

  const int lane = tid & 31;
  const int wid  = tid >> 5;
  const float* __restrict__ tgtbase =
      htgt + ((size_t)(b * SS * LL + l)) * HH;  // + t*LL*HH per row

  for (int t = wid; t < SS; t += 8) {
    const float* __restrict__ trow = tgtbase + (size_t)t * (LL * HH);
    float acc = 0.0f;
#pragma unroll 4
    for (int h = lane; h < HH; h += 32) {
      float x = s_src[h] + trow[h];
      // exact GELU: 0.5*x*(1+erf(x/sqrt(2)))
      float g = 0.5f * x * (1.0f + erff(x * 0.70710678118654752f));
      acc = fmaf(g, s_w[h], acc);
    }
#pragma unroll
    for (int off = 16; off > 0; off >>= 1) acc += __shfl_xor(acc, off, 32);
    if (lane == 0) {
      out[((size_t)((b * SS + s) * SS + t)) * LL + l] = acc;
    }
  }
}

extern "C" void kernel_launch(void* const* d_in, const int* in_sizes, int n_in,
                              void* d_out, int out_size, void* d_ws, size_t ws_size,
                              hipStream_t stream) {
  const float* pooled = (const float*)d_in[0];
  const float* W_src  = (const float*)d_in[1];
  const float* b_src  = (const float*)d_in[2];
  const float* W_tgt  = (const float*)d_in[3];
  const float* b_tgt  = (const float*)d_in[4];
  const float* wcls   = (const float*)d_in[5];
  float* out = (float*)d_out;

  float* hsrc = (float*)d_ws;
  float* htgt = hsrc + (size_t)MM * NN;

  dim3 g1(NN / 16, MM / 16);  // 192 x 24 tiles, one wave each
  proj_wmma_f32<<<g1, 32, 0, stream>>>(pooled, W_src, b_src, hsrc);
  proj_wmma_f32<<<g1, 32, 0, stream>>>(pooled, W_tgt, b_tgt, htgt);

  dim3 g2(SS, LL, BB);  // (s, l, b)
  pair_gelu_reduce<<<g2, 256, 0, stream>>>(hsrc, htgt, wcls, out);
}